// Multi_Htrans_Layer_62904091018242
// MI455X (gfx1250) — compile-verified
//
#include <hip/hip_runtime.h>
#include <math.h>

// ---- problem constants (from reference) ----
#define E_N   20000
#define R_N   200
#define NE_N  100000
#define H_N   4
#define IN_N  300
#define DK_N  75
#define DK2   150          // 2*DK
#define DK3   225          // 3*DK
#define AG_STRIDE 228      // DK3 padded to multiple of 4 (WMMA K step), zero-filled
#define KSPLIT 25          // K-split factor for the [200,20000]@[20000,75] GEMM
#define KCHUNK (E_N / KSPLIT)   // 800
#define ALPHA_F 0.2f
#define INV_SQRT_DK 0.1154700538379252f  // 1/sqrt(75)

typedef __attribute__((ext_vector_type(2))) float v2f;
typedef __attribute__((ext_vector_type(8))) float v8f;

// D = A(16x4 f32) * B(4x16 f32) + C(16x16 f32), wave32 WMMA
__device__ __forceinline__ v8f wmma_f32_4(v2f a, v2f b, v8f c) {
  return __builtin_amdgcn_wmma_f32_16x16x4_f32(
      /*neg_a=*/false, a, /*neg_b=*/false, b,
      /*c_mod=*/(short)0, c, /*reuse_a=*/false, /*reuse_b=*/false);
}

// ============================================================
// Stage 1: node projections.  For g = mat*4 + h (mat: 0=K,1=Q,2=V,3=XL,4=XR)
// P[g] = x @ W[mat][h] (+ bias for mat<3).  One block = 16 rows of x staged
// in LDS, 5 waves each own one 16-col N tile (75 -> 5x16 padded).
// Out-of-tile lanes use a CLAMPED column index (branch-free loads); their
// results land in dead C columns which are never stored.
// ============================================================
__global__ __launch_bounds__(160)
void k_proj(const float* __restrict__ x,
            const float* __restrict__ Wk, const float* __restrict__ bk,
            const float* __restrict__ Wq, const float* __restrict__ bq,
            const float* __restrict__ Wv, const float* __restrict__ bv,
            const float* __restrict__ rl, const float* __restrict__ rr,
            float* __restrict__ proj)
{
  __shared__ float As[16 * IN_N];                 // 19.2 KB
  const int m0  = blockIdx.x * 16;
  const int g   = blockIdx.y;                     // 0..19
  const int mat = g >> 2, h = g & 3;

  const float* W;  const float* bias = nullptr;
  switch (mat) {
    case 0: W = Wk + (size_t)h * IN_N * DK_N; bias = bk + h * DK_N; break;
    case 1: W = Wq + (size_t)h * IN_N * DK_N; bias = bq + h * DK_N; break;
    case 2: W = Wv + (size_t)h * IN_N * DK_N; bias = bv + h * DK_N; break;
    case 3: W = rl + (size_t)h * IN_N * DK_N; break;
    default:W = rr + (size_t)h * IN_N * DK_N; break;
  }
  float* P = proj + (size_t)g * ((size_t)E_N * DK_N);

  // 16 consecutive rows of x are contiguous -> one flat coalesced copy
  const float* src = x + (size_t)m0 * IN_N;
  for (int i = threadIdx.x; i < 16 * IN_N; i += 160) As[i] = src[i];
  __syncthreads();

  const int lane = threadIdx.x & 31;
  const int wv   = threadIdx.x >> 5;      // wave id 0..4 -> N tile
  const int l16  = lane & 15;
  const int hi   = lane >> 4;
  const int koff = hi * 2;                // A/B frag K sub-offset per half-wave
  const int n    = wv * 16 + l16;
  const bool nok = (n < DK_N);
  const int ncl  = nok ? n : (DK_N - 1);  // clamped (branch-free loads)

  v8f c = {};
  #pragma unroll 5
  for (int k = 0; k < IN_N; k += 4) {
    v2f a = *reinterpret_cast<const v2f*>(&As[l16 * IN_N + k + koff]);
    v2f b;
    b.x = W[(k + koff)     * DK_N + ncl];
    b.y = W[(k + koff + 1) * DK_N + ncl];
    c = wmma_f32_4(a, b, c);
  }
  if (nok) {
    const float bb = bias ? bias[n] : 0.f;
    for (int v = 0; v < 8; ++v) {
      const int r = m0 + v + hi * 8;
      P[(size_t)r * DK_N + n] = c[v] + bb;
    }
  }
}

// ============================================================
// Stage 2a: inverse row sums of r_head / r_tail (hs, ts)
// ============================================================
__global__ void k_rsum(const float* __restrict__ rh, const float* __restrict__ rt,
                       float* __restrict__ rsumi)
{
  const int mat = blockIdx.y;            // 0=head, 1=tail
  const int r   = blockIdx.x;
  const float* base = (mat == 0 ? rh : rt) + (size_t)r * E_N;
  float s = 0.f;
  for (int i = threadIdx.x; i < E_N; i += blockDim.x) s += base[i];
  __shared__ float red[256];
  red[threadIdx.x] = s; __syncthreads();
  for (int o = 128; o > 0; o >>= 1) {
    if ((int)threadIdx.x < o) red[threadIdx.x] += red[threadIdx.x + o];
    __syncthreads();
  }
  if (threadIdx.x == 0) {
    const float t = red[0];
    rsumi[mat * R_N + r] = (t == 0.f) ? 0.f : 1.f / t;
  }
}

// ============================================================
// Stage 2b: Lraw = r_head @ (x@rel_L), Rraw = r_tail @ (x@rel_R)
// [200,20000]@[20000,75] via WMMA.  One wave owns a 16-row M tile and ALL
// 5 N tiles (A fragment reused across 5 WMMAs).  K is split KSPLIT ways
// across blocks; partials reduced with f32 atomics into zeroed Lraw.
// Clamped indices keep all loads branch-free.
// ============================================================
__global__ __launch_bounds__(32)
void k_rel_gemm(const float* __restrict__ rh, const float* __restrict__ rt,
                const float* __restrict__ xlxr, float* __restrict__ lraw)
{
  const int m0  = blockIdx.x * 16;       // 13 tiles (200 pad->208)
  const int g   = blockIdx.y;            // mat*4 + h, mat 0=L (head), 1=R (tail)
  const int kc0 = blockIdx.z * KCHUNK;   // K split
  const int mat = g >> 2;
  const float* A = (mat == 0) ? rh : rt;                       // [200, 20000]
  const float* B = xlxr + (size_t)g * ((size_t)E_N * DK_N);    // [20000, 75]

  const int lane = threadIdx.x;
  const int l16 = lane & 15, hi = lane >> 4, koff = hi * 2;
  const int r  = m0 + l16;
  const int rc = (r < R_N) ? r : (R_N - 1);        // clamp dead A rows
  const int n4  = 64 + l16;                        // last tile columns
  const int n4c = (n4 < DK_N) ? n4 : (DK_N - 1);   // clamp dead B cols

  v8f c0 = {}, c1 = {}, c2 = {}, c3 = {}, c4 = {};
  for (int k = kc0; k < kc0 + KCHUNK; k += 4) {
    const v2f a = *reinterpret_cast<const v2f*>(&A[(size_t)rc * E_N + k + koff]);
    const float* Brow0 = B + (size_t)(k + koff) * DK_N;
    const float* Brow1 = Brow0 + DK_N;
    v2f b;
    b.x = Brow0[l16];        b.y = Brow1[l16];        c0 = wmma_f32_4(a, b, c0);
    b.x = Brow0[16 + l16];   b.y = Brow1[16 + l16];   c1 = wmma_f32_4(a, b, c1);
    b.x = Brow0[32 + l16];   b.y = Brow1[32 + l16];   c2 = wmma_f32_4(a, b, c2);
    b.x = Brow0[48 + l16];   b.y = Brow1[48 + l16];   c3 = wmma_f32_4(a, b, c3);
    b.x = Brow0[n4c - l16 + l16]; b.y = Brow1[n4c];   // clamped last tile
    b.x = Brow0[n4c];
    c4 = wmma_f32_4(a, b, c4);
  }

  float* Lg = lraw + (size_t)g * R_N * DK_N;
  for (int v = 0; v < 8; ++v) {
    const int rr_ = m0 + v + hi * 8;
    if (rr_ < R_N) {
      float* row = Lg + (size_t)rr_ * DK_N;
      __hip_atomic_fetch_add(&row[l16],      c0[v], __ATOMIC_RELAXED, __HIP_MEMORY_SCOPE_AGENT);
      __hip_atomic_fetch_add(&row[16 + l16], c1[v], __ATOMIC_RELAXED, __HIP_MEMORY_SCOPE_AGENT);
      __hip_atomic_fetch_add(&row[32 + l16], c2[v], __ATOMIC_RELAXED, __HIP_MEMORY_SCOPE_AGENT);
      __hip_atomic_fetch_add(&row[48 + l16], c3[v], __ATOMIC_RELAXED, __HIP_MEMORY_SCOPE_AGENT);
      if (n4 < DK_N)
        __hip_atomic_fetch_add(&row[n4], c4[v], __ATOMIC_RELAXED, __HIP_MEMORY_SCOPE_AGENT);
    }
  }
}

// ============================================================
// Stage 2c: rel_embed = relu([L*hs, R*ts]); rkq = rel_embed * attr_r
// ============================================================
__global__ void k_relfin(const float* __restrict__ lraw, const float* __restrict__ rsumi,
                         const float* __restrict__ attr, float* __restrict__ rele,
                         float* __restrict__ rkq)
{
  const int idx = blockIdx.x * blockDim.x + threadIdx.x;   // H*R*150
  if (idx >= H_N * R_N * DK2) return;
  const int d = idx % DK2;
  const int t = idx / DK2;
  const int r = t % R_N;
  const int h = t / R_N;
  const int mat = (d < DK_N) ? 0 : 1;
  const int dd  = (mat ? d - DK_N : d);
  float v = lraw[(((size_t)mat * H_N + h) * R_N + r) * DK_N + dd] * rsumi[mat * R_N + r];
  v = v > 0.f ? v : 0.f;
  rele[((size_t)h * R_N + r) * DK2 + d] = v;
  rkq [((size_t)h * R_N + r) * DK2 + d] = v * attr[h * DK2 + d];
}

__global__ void k_zero(float* __restrict__ p, int n) {
  const int i = blockIdx.x * blockDim.x + threadIdx.x;
  if (i < n) p[i] = 0.f;
}

// order-preserving float->uint encoding for atomicMax over floats
__device__ __forceinline__ unsigned enc_f(float x) {
  const unsigned b = __float_as_uint(x);
  return (b & 0x80000000u) ? ~b : (b | 0x80000000u);
}
__device__ __forceinline__ float dec_f(unsigned u) {
  const unsigned b = (u & 0x80000000u) ? (u & 0x7FFFFFFFu) : ~u;
  return __uint_as_float(b);
}

// ============================================================
// Stage 3a: per-(edge,head) attention logit + segment max.
// 16 lanes cooperate on the two 75-length dots.
// ============================================================
__global__ void k_attA(const int* __restrict__ ei, const int* __restrict__ et,
                       const float* __restrict__ Kb, const float* __restrict__ Qb,
                       const float* __restrict__ rkq, float* __restrict__ att,
                       unsigned* __restrict__ mxu)
{
  const int unit = blockIdx.x * (blockDim.x >> 4) + ((int)threadIdx.x >> 4);
  const int l = threadIdx.x & 15;
  if (unit >= NE_N * H_N) return;
  const int e = unit >> 2, h = unit & 3;
  const int src = ei[e], tgt = ei[NE_N + e], ty = et[e];
  const float* kk = Kb + ((size_t)h * E_N + src) * DK_N;
  const float* qq = Qb + ((size_t)h * E_N + tgt) * DK_N;
  const float* rv = rkq + ((size_t)h * R_N + ty) * DK2;
  float s = 0.f;
  for (int d = l; d < DK_N; d += 16)
    s += kk[d] * rv[d] + qq[d] * rv[DK_N + d];
  s += __shfl_xor(s, 8, 16);
  s += __shfl_xor(s, 4, 16);
  s += __shfl_xor(s, 2, 16);
  s += __shfl_xor(s, 1, 16);
  if (l == 0) {
    float a = s * INV_SQRT_DK;
    a = (a > 0.f) ? a : ALPHA_F * a;   // leaky_relu
    a = -a;
    att[(size_t)h * NE_N + e] = a;
    atomicMax(&mxu[h * E_N + tgt], enc_f(a));
  }
}

// ============================================================
// Stage 3b: e = exp(att - max); segment sum
// ============================================================
__global__ void k_attB(const int* __restrict__ ei, const unsigned* __restrict__ mxu,
                       float* __restrict__ att, float* __restrict__ ssum)
{
  const int i = blockIdx.x * blockDim.x + threadIdx.x;
  if (i >= NE_N * H_N) return;
  const int e = i % NE_N, h = i / NE_N;          // att laid out [h][e]
  const int tgt = ei[NE_N + e];
  const float m = dec_f(mxu[h * E_N + tgt]);
  const float ex = __expf(att[i] - m);
  att[i] = ex;
  __hip_atomic_fetch_add(&ssum[h * E_N + tgt], ex,
                         __ATOMIC_RELAXED, __HIP_MEMORY_SCOPE_AGENT);
}

// ============================================================
// Stage 3c: normalized weighted scatter-add of msg = [V[src], rel_embed[type]]
// one wave per (edge, head); 225 columns strided over 32 lanes.
// ============================================================
__global__ void k_attC(const int* __restrict__ ei, const int* __restrict__ et,
                       const float* __restrict__ att, const float* __restrict__ ssum,
                       const float* __restrict__ Vb, const float* __restrict__ rele,
                       float* __restrict__ aggr)
{
  const int unit = blockIdx.x * (blockDim.x >> 5) + ((int)threadIdx.x >> 5);
  const int lane = threadIdx.x & 31;
  if (unit >= NE_N * H_N) return;
  const int e = unit >> 2, h = unit & 3;
  const int src = ei[e], tgt = ei[NE_N + e], ty = et[e];
  const float w = att[(size_t)h * NE_N + e] / (ssum[h * E_N + tgt] + 1e-16f);
  const float* vv = Vb + ((size_t)h * E_N + src) * DK_N;
  const float* re = rele + ((size_t)h * R_N + ty) * DK2;
  float* ag = aggr + ((size_t)h * E_N + tgt) * AG_STRIDE;
  for (int d = lane; d < DK3; d += 32) {
    const float m = (d < DK_N) ? vv[d] : re[d - DK_N];
    __hip_atomic_fetch_add(&ag[d], w * m,
                           __ATOMIC_RELAXED, __HIP_MEMORY_SCOPE_AGENT);
  }
}

// ============================================================
// Stage 4: out = (relu(aggr) @ Wo + bo) * sigmoid(skw) + emb * (1-sigmoid(skw))
// WMMA, K = 228 (225 zero-padded).  One wave per 16-row M tile, 5 N tiles
// with A-fragment reuse.  K tail (kc>=225) multiplies clamped-but-finite W
// values by exact 0.0 from the zero pad, contributing nothing.
// ============================================================
__global__ __launch_bounds__(32)
void k_out(const float* __restrict__ aggr, const float* __restrict__ Wo,
           const float* __restrict__ bo, const float* __restrict__ emb,
           const float* __restrict__ skw, float* __restrict__ out)
{
  const int m0 = blockIdx.x * 16;        // 1250 tiles
  const int h  = blockIdx.y;
  const float* A = aggr + (size_t)h * E_N * AG_STRIDE;   // [20000, 228]
  const float* W = Wo   + (size_t)h * DK3 * DK_N;        // [225, 75]

  const int lane = threadIdx.x;
  const int l16 = lane & 15, hi = lane >> 4, koff = hi * 2;
  const int n4  = 64 + l16;
  const int n4c = (n4 < DK_N) ? n4 : (DK_N - 1);

  v8f c0 = {}, c1 = {}, c2 = {}, c3 = {}, c4 = {};
  for (int k = 0; k < AG_STRIDE; k += 4) {
    v2f a = *reinterpret_cast<const v2f*>(&A[(size_t)(m0 + l16) * AG_STRIDE + k + koff]);
    a.x = a.x > 0.f ? a.x : 0.f;         // relu fused into A fragment
    a.y = a.y > 0.f ? a.y : 0.f;
    const int kc0c = (k + koff     < DK3) ? (k + koff)     : (DK3 - 1);  // clamp K tail
    const int kc1c = (k + koff + 1 < DK3) ? (k + koff + 1) : (DK3 - 1);
    const float* Wr0 = W + (size_t)kc0c * DK_N;
    const float* Wr1 = W + (size_t)kc1c * DK_N;
    v2f b;
    b.x = Wr0[l16];      b.y = Wr1[l16];      c0 = wmma_f32_4(a, b, c0);
    b.x = Wr0[16 + l16]; b.y = Wr1[16 + l16]; c1 = wmma_f32_4(a, b, c1);
    b.x = Wr0[32 + l16]; b.y = Wr1[32 + l16]; c2 = wmma_f32_4(a, b, c2);
    b.x = Wr0[48 + l16]; b.y = Wr1[48 + l16]; c3 = wmma_f32_4(a, b, c3);
    b.x = Wr0[n4c];      b.y = Wr1[n4c];      c4 = wmma_f32_4(a, b, c4);
  }

  const float alpha = 1.f / (1.f + __expf(-skw[0]));
  for (int v = 0; v < 8; ++v) {
    const int r = m0 + v + hi * 8;
    const size_t rowo = (size_t)r * IN_N + (size_t)h * DK_N;
    {
      const float ho = c0[v] + bo[h * DK_N + l16];
      out[rowo + l16] = ho * alpha + emb[rowo + l16] * (1.f - alpha);
    }
    {
      const float ho = c1[v] + bo[h * DK_N + 16 + l16];
      out[rowo + 16 + l16] = ho * alpha + emb[rowo + 16 + l16] * (1.f - alpha);
    }
    {
      const float ho = c2[v] + bo[h * DK_N + 32 + l16];
      out[rowo + 32 + l16] = ho * alpha + emb[rowo + 32 + l16] * (1.f - alpha);
    }
    {
      const float ho = c3[v] + bo[h * DK_N + 48 + l16];
      out[rowo + 48 + l16] = ho * alpha + emb[rowo + 48 + l16] * (1.f - alpha);
    }
    if (n4 < DK_N) {
      const float ho = c4[v] + bo[h * DK_N + n4];
      out[rowo + n4] = ho * alpha + emb[rowo + n4] * (1.f - alpha);
    }
  }
}

// ============================================================
extern "C" void kernel_launch(void* const* d_in, const int* in_sizes, int n_in,
                              void* d_out, int out_size, void* d_ws, size_t ws_size,
                              hipStream_t stream)
{
  const float* x   = (const float*)d_in[0];
  const float* emb = (const float*)d_in[1];
  const int*   ei  = (const int*)  d_in[2];
  const int*   et  = (const int*)  d_in[3];
  const float* rh  = (const float*)d_in[4];
  const float* rt  = (const float*)d_in[5];
  const float* Wk  = (const float*)d_in[6];
  const float* bk  = (const float*)d_in[7];
  const float* Wq  = (const float*)d_in[8];
  const float* bq  = (const float*)d_in[9];
  const float* Wv  = (const float*)d_in[10];
  const float* bv  = (const float*)d_in[11];
  const float* Wo  = (const float*)d_in[12];
  const float* bo  = (const float*)d_in[13];
  const float* ar  = (const float*)d_in[14];
  const float* rl  = (const float*)d_in[15];
  const float* rr  = (const float*)d_in[16];
  const float* skw = (const float*)d_in[17];
  float* out = (float*)d_out;
  float* ws  = (float*)d_ws;

  // ---- workspace layout (~149 MB) ----
  const size_t PER = (size_t)H_N * E_N * DK_N;       // 6,000,000 floats
  float* PROJ = ws;                                  // [5][H][E][75]
  float* Kb   = PROJ + 0 * PER;
  float* Qb   = PROJ + 1 * PER;
  float* Vb   = PROJ + 2 * PER;
  float* XLXR = PROJ + 3 * PER;                      // [2][H][E][75]
  float* AGGR = XLXR;                                // aliases XL/XR after k_rel_gemm
  const size_t AGGR_ELEMS = (size_t)H_N * E_N * AG_STRIDE;   // 18,240,000
  float* LRAW  = AGGR + AGGR_ELEMS;                  // [2][H][200][75]
  const int LRAW_ELEMS = 2 * H_N * R_N * DK_N;       // 120,000
  float* RELE  = LRAW + (size_t)LRAW_ELEMS;
  float* RKQ   = RELE + (size_t)H_N * R_N * DK2;
  float* RSUMI = RKQ  + (size_t)H_N * R_N * DK2;
  float* ATT   = RSUMI + 2 * R_N;                    // [H][NE]
  float* SSUM  = ATT + (size_t)H_N * NE_N;           // [H][E]
  unsigned* MXU = (unsigned*)(SSUM + (size_t)H_N * E_N);  // [H][E], contiguous after SSUM

  // 1) node projections (K,Q,V, x@rel_L, x@rel_R) -- WMMA
  k_proj<<<dim3(E_N / 16, 20), 160, 0, stream>>>(x, Wk, bk, Wq, bq, Wv, bv, rl, rr, PROJ);

  // 2) relation embeddings (K-split WMMA + atomic reduce into zeroed Lraw)
  k_rsum<<<dim3(R_N, 2), 256, 0, stream>>>(rh, rt, RSUMI);
  k_zero<<<(LRAW_ELEMS + 255) / 256, 256, 0, stream>>>(LRAW, LRAW_ELEMS);
  k_rel_gemm<<<dim3(13, 8, KSPLIT), 32, 0, stream>>>(rh, rt, XLXR, LRAW);
  {
    const int n = H_N * R_N * DK2;
    k_relfin<<<(n + 255) / 256, 256, 0, stream>>>(LRAW, RSUMI, ar, RELE, RKQ);
  }

  // 3) init accumulators (AGGR overwrites dead XL/XR region; SSUM+MXU contiguous)
  k_zero<<<((int)AGGR_ELEMS + 255) / 256, 256, 0, stream>>>(AGGR, (int)AGGR_ELEMS);
  k_zero<<<(2 * H_N * E_N + 255) / 256, 256, 0, stream>>>(SSUM, 2 * H_N * E_N);

  // 4) attention: logits + segment max, exp + segment sum, weighted scatter-add
  k_attA<<<(NE_N * H_N) / 16, 256, 0, stream>>>(ei, et, Kb, Qb, RKQ, ATT, MXU);
  k_attB<<<(NE_N * H_N + 255) / 256, 256, 0, stream>>>(ei, MXU, ATT, SSUM);
  k_attC<<<(NE_N * H_N) / 8, 256, 0, stream>>>(ei, et, ATT, SSUM, Vb, RELE, AGGR);

  // 5) output GEMM + skip blend -- WMMA
  k_out<<<dim3(E_N / 16, H_N), 32, 0, stream>>>(AGGR, Wo, bo, emb, skw, out);
}